// GSDepthRankingLoss_11304353923620
// MI455X (gfx1250) — compile-verified
//
#include <hip/hip_runtime.h>
#include <stdint.h>

typedef __attribute__((ext_vector_type(2))) float v2f;
typedef __attribute__((ext_vector_type(8))) float v8f;

#define PAD_VAL   (-1000000.0f)
#define RADIUS    3
#define WIN       7
#define WIN2      49
#define RANK_M    1.0e-4f
#define CONT_M    1.0e-4f
#define WEIGHT    0.2f
#define CONT_W    0.1f

// ---------------- Threefry2x32-20 (JAX core hash), deterministic per sample --
__device__ __forceinline__ uint32_t rotl32(uint32_t x, int r) {
    return (x << r) | (x >> (32 - r));
}

__device__ __forceinline__ void threefry2x32(uint32_t k0, uint32_t k1,
                                             uint32_t c0, uint32_t c1,
                                             uint32_t& o0, uint32_t& o1) {
    const uint32_t ks2 = k0 ^ k1 ^ 0x1BD11BDAu;
    uint32_t x0 = c0 + k0, x1 = c1 + k1;
    const int RA[4] = {13, 15, 26, 6};
    const int RB[4] = {17, 29, 16, 24};
#pragma unroll
    for (int i = 0; i < 4; ++i) { x0 += x1; x1 = rotl32(x1, RA[i]); x1 ^= x0; }
    x0 += k1;  x1 += ks2 + 1u;
#pragma unroll
    for (int i = 0; i < 4; ++i) { x0 += x1; x1 = rotl32(x1, RB[i]); x1 ^= x0; }
    x0 += ks2; x1 += k0 + 2u;
#pragma unroll
    for (int i = 0; i < 4; ++i) { x0 += x1; x1 = rotl32(x1, RA[i]); x1 ^= x0; }
    x0 += k0;  x1 += k1 + 3u;
#pragma unroll
    for (int i = 0; i < 4; ++i) { x0 += x1; x1 = rotl32(x1, RB[i]); x1 ^= x0; }
    x0 += k1;  x1 += ks2 + 4u;
#pragma unroll
    for (int i = 0; i < 4; ++i) { x0 += x1; x1 = rotl32(x1, RA[i]); x1 ^= x0; }
    o0 = x0 + ks2;
    o1 = x1 + k0 + 5u;
}

// ---------------- init ws accumulators --------------------------------------
__global__ void gs_rank_init(float* __restrict__ acc) {
    if (threadIdx.x < 4) acc[threadIdx.x] = 0.0f;
}

// ---------------- main: sample, crop-select, loss terms, WMMA reduction -----
__global__ __launch_bounds__(256) void gs_rank_main(
    const float* __restrict__ td,   // target_depths (H*W)
    const float* __restrict__ rd,   // render_depths (H*W)
    const int*   __restrict__ vm,   // valid_mask    (H*W)
    const int*   __restrict__ pw,   // width  scalar
    const int*   __restrict__ ph,   // height scalar
    float*       __restrict__ acc)  // [rank_sum, cont_sum, mask_sum]
{
    const int W = *pw;
    const int H = *ph;
    const int n = (W * H) >> 2;          // SAMPLE_RATIO = 0.25
    const int patch = W >> 3;            // rank_patch_size = width // 8
    const int ymax = H - patch;
    const int xmax = W - patch;

    float rank_acc = 0.0f, cont_acc = 0.0f, mask_acc = 0.0f;

    const int stride = (int)(gridDim.x * blockDim.x);
    for (int i = (int)(blockIdx.x * blockDim.x + threadIdx.x); i < n; i += stride) {
        // ---- per-sample randoms (Threefry counter mode) ----
        uint32_t a0, a1, b0, b1, c0, c1;
        threefry2x32(42u, 0x9E3779B9u, (uint32_t)i, 0u, a0, a1);
        threefry2x32(42u, 0x9E3779B9u, (uint32_t)i, 1u, b0, b1);
        threefry2x32(42u, 0x9E3779B9u, (uint32_t)i, 2u, c0, c1);

        const int syb = (int)(a0 % (uint32_t)ymax);
        const int sxb = (int)(a1 % (uint32_t)xmax);

        int sy[2], sx[2], nbr[2], sidx[2], nidx[2];
        float sd[2];
        sy[0] = syb + (int)((b0 & 0xFFFFu) % (uint32_t)patch);
        sy[1] = syb + (int)((b0 >> 16)     % (uint32_t)patch);
        sx[0] = sxb + (int)((b1 & 0xFFFFu) % (uint32_t)patch);
        sx[1] = sxb + (int)((b1 >> 16)     % (uint32_t)patch);
        nbr[0] = 1 + (int)(c0 % 14u);    // top_n = 15 -> nbr in [1,14]
        nbr[1] = 1 + (int)(c1 % 14u);

#pragma unroll
        for (int j = 0; j < 2; ++j) {
            sidx[j] = sy[j] * W + sx[j];
            sd[j]   = td[sidx[j]];
        }

        // ---- 7x7 crop distances + stable k-th-smallest selection ----
#pragma unroll
        for (int j = 0; j < 2; ++j) {
            float a[WIN2];
#pragma unroll
            for (int t = 0; t < WIN2; ++t) {
                const int yy = sy[j] + (t / WIN) - RADIUS;
                const int xx = sx[j] + (t % WIN) - RADIUS;
                const bool inb = (yy >= 0) & (yy < H) & (xx >= 0) & (xx < W);
                const float v = inb ? td[yy * W + xx] : PAD_VAL;
                a[t] = fabsf(v - sd[j]);
            }
            // iterated lexicographic-min == stable argsort selection
            float curV = -1.0f;
            int   curI = -1;
            const int k = nbr[j];
            for (int p = 0; p <= k; ++p) {
                float bv = 3.0e38f;
                int   bi = WIN2;
#pragma unroll
                for (int t = 0; t < WIN2; ++t) {
                    const bool gt = (a[t] > curV) || ((a[t] == curV) && (t > curI));
                    const bool lt = (a[t] < bv)  || ((a[t] == bv)  && (t < bi));
                    if (gt && lt) { bv = a[t]; bi = t; }
                }
                curV = bv; curI = bi;
            }
            const int ny = sy[j] - RADIUS + curI / WIN;
            const int nx = sx[j] - RADIUS + curI % WIN;
            nidx[j] = ny * W + nx;      // always in-bounds: pads sort last
        }

        // ---- validity mask (channel dim == 1 -> any() is !=0) ----
        const bool mOK = (vm[sidx[0]] != 0) & (vm[sidx[1]] != 0) &
                         (vm[nidx[0]] != 0) & (vm[nidx[1]] != 0);

        // ---- descending stable order by sampled target depth ----
        int o0 = 0, o1 = 1;
        if (sd[1] > sd[0]) { o0 = 1; o1 = 0; }

        const float d00 = rd[sidx[o0]];
        const float d01 = rd[sidx[o1]];
        const float d10 = rd[nidx[o0]];
        const float d11 = rd[nidx[o1]];

        const float rank = fmaxf(d00 - d01 + RANK_M, 0.0f);
        const float cont = fmaxf(fabsf(d00 - d10) - CONT_M, 0.0f) +
                           fmaxf(fabsf(d01 - d11) - CONT_M, 0.0f);

        const float fm = mOK ? 1.0f : 0.0f;
        rank_acc += rank * fm;
        cont_acc += cont * fm;
        mask_acc += fm;
    }

    // ---- wave32 reduction via V_WMMA_F32_16X16X4_F32 with B = ones ----------
    // A layout (16x4 f32): lanes 0-15 hold row M=lane (K0,K1 in v0,v1),
    // lanes 16-31 hold row M=lane-16 (K2,K3). With B all-ones,
    // D[m][n] = rowsum(m) = val[lane m] + val[lane m+16] (K1/K3 slots = 0).
    // Lane n<16 sums D rows 0..7; lane n>=16 rows 8..15; shfl_xor(16) totals.
    v2f ones; ones.x = 1.0f; ones.y = 1.0f;
    v8f cz = {};
    float vals[3] = {rank_acc, cont_acc, mask_acc};
#pragma unroll
    for (int q = 0; q < 3; ++q) {
        v2f av; av.x = vals[q]; av.y = 0.0f;
        v8f dres = __builtin_amdgcn_wmma_f32_16x16x4_f32(
            /*neg_a=*/false, av, /*neg_b=*/false, ones,
            /*c_mod=*/(short)0, cz, /*reuse_a=*/false, /*reuse_b=*/false);
        float s = dres[0] + dres[1] + dres[2] + dres[3] +
                  dres[4] + dres[5] + dres[6] + dres[7];
        s += __shfl_xor(s, 16, 32);
        vals[q] = s;
    }
    if ((threadIdx.x & 31u) == 0u) {
        atomicAdd(&acc[0], vals[0]);
        atomicAdd(&acc[1], vals[1]);
        atomicAdd(&acc[2], vals[2]);
    }
}

// ---------------- finalize ---------------------------------------------------
__global__ void gs_rank_final(const float* __restrict__ acc, float* __restrict__ out) {
    const float denom = fmaxf(acc[2], 1.0f);
    out[0] = WEIGHT * (acc[0] / denom);
    out[1] = WEIGHT * CONT_W * (acc[1] / (denom * 2.0f));
}

extern "C" void kernel_launch(void* const* d_in, const int* in_sizes, int n_in,
                              void* d_out, int out_size, void* d_ws, size_t ws_size,
                              hipStream_t stream) {
    const float* td = (const float*)d_in[0];   // target_depths (1, H*W) f32
    const float* rd = (const float*)d_in[1];   // render_depths (H*W)    f32
    const int*   vm = (const int*)  d_in[2];   // valid_mask    (1,H*W,1) i32
    const int*   pw = (const int*)  d_in[3];   // width  scalar
    const int*   ph = (const int*)  d_in[4];   // height scalar
    float* out = (float*)d_out;                // 2 floats
    float* acc = (float*)d_ws;                 // 3 accumulators + pad

    gs_rank_init <<<1, 32, 0, stream>>>(acc);
    gs_rank_main <<<2048, 256, 0, stream>>>(td, rd, vm, pw, ph, acc);
    gs_rank_final<<<1, 1, 0, stream>>>(acc, out);
}